// SinkhornDistance_21955872817205
// MI455X (gfx1250) — compile-verified
//
#include <hip/hip_runtime.h>
#include <math.h>

// Sinkhorn distance, N=4096, M=512, D=32, EPS=2, THRESH=0.1, MAX_ITER=100.
// Per-row closed-form reduction: only A=ΣK, S=ΣKr, T=ΣKrC per row are needed,
// then a scalar fixed-point iteration with a global early-stop reduction.

typedef __attribute__((ext_vector_type(2))) float v2f;
typedef __attribute__((ext_vector_type(8))) float v8f;

#define N_ 4096
#define M_ 512
#define D_ 32
#define EPS_ 2.0f
#define THRESH_ 0.1f
#define MAX_ITER_ 100
#define MCHUNKS_ 4              // split M into 4 chunks of 128 cols for occupancy
#define MTILES_PER_CHUNK_ 8     // 8 tiles of 16 cols each

// ---------------------------------------------------------------- kernel 0
// squared norms of x rows and y rows
__global__ void sq_norms_kernel(const float* __restrict__ x,
                                const float* __restrict__ y,
                                float* __restrict__ xsq,
                                float* __restrict__ ysq) {
  int i = blockIdx.x * blockDim.x + threadIdx.x;
  if (i < N_) {
    const float* p = x + i * D_;
    float s = 0.f;
#pragma unroll
    for (int k = 0; k < D_; ++k) s += p[k] * p[k];
    xsq[i] = s;
  }
  if (i < M_) {
    const float* p = y + i * D_;
    float s = 0.f;
#pragma unroll
    for (int k = 0; k < D_; ++k) s += p[k] * p[k];
    ysq[i] = s;
  }
}

// ---------------------------------------------------------------- kernel 1
// One wave per (16-row x-tile, 128-col m-chunk).
// Gram tile G = x_tile . y_tile^T via V_WMMA_F32_16X16X4_F32 (full fp32),
// then fused C = sqrt(xsq+ysq-2G), K = exp(-C/EPS).
// Per-lane partial sums over all 8 m-tiles; single 16-lane xor reduction at end.
// launch_bounds(256,1): allow up to 256 VGPRs so the fully-unrolled 8-tile
// WMMA pipeline stays spill-free (memory-bound kernel; occupancy is secondary).
__global__ void __launch_bounds__(256, 1)
gram_rowsum_kernel(const float* __restrict__ x,
                   const float* __restrict__ y,
                   const float* __restrict__ r,
                   const float* __restrict__ xsq,
                   const float* __restrict__ ysq,
                   float* __restrict__ partK,    // [MCHUNKS_][N_]
                   float* __restrict__ partKr,   // [MCHUNKS_][N_]
                   float* __restrict__ partKrC)  // [MCHUNKS_][N_]
{
  const int lane  = threadIdx.x & 31;
  const int wid   = threadIdx.x >> 5;           // 8 waves per block
  const int half  = lane >> 4;                  // 0: lanes 0-15, 1: lanes 16-31
  const int lq    = lane & 15;
  const int ntile = blockIdx.x * 8 + wid;       // 0..255
  const int n0    = ntile * 16;
  const int chunk = blockIdx.y;                 // 0..3
  const int mbase = chunk * MTILES_PER_CHUNK_ * 16;

  // A fragment (x tile), hoisted: 16x4 f32 per K-step.
  // Lane layout per ISA: M = lane%16; VGPR0/1 hold K = 4*ks + 2*half + {0,1}.
  v2f a[8];
#pragma unroll
  for (int ks = 0; ks < 8; ++ks) {
    const float* p = x + (n0 + lq) * D_ + ks * 4 + 2 * half;
    a[ks].x = p[0];
    a[ks].y = p[1];
  }

  // xsq for the 8 rows this lane's D-fragment covers: row = n0 + v + 8*half
  float xsv[8];
#pragma unroll
  for (int v = 0; v < 8; ++v) xsv[v] = xsq[n0 + v + 8 * half];

  // per-lane partial sums (this lane's column subset of each row)
  float accK[8], accKr[8], accKrC[8];
#pragma unroll
  for (int v = 0; v < 8; ++v) { accK[v] = 0.f; accKr[v] = 0.f; accKrC[v] = 0.f; }

#pragma unroll
  for (int mt = 0; mt < MTILES_PER_CHUNK_; ++mt) {
    const int m0 = mbase + mt * 16;

    // B fragments (y tile) first: one load clause, then uninterrupted WMMA chain
    v2f b[8];
#pragma unroll
    for (int ks = 0; ks < 8; ++ks) {
      const float* p = y + (m0 + lq) * D_ + ks * 4 + 2 * half;
      b[ks].x = p[0];
      b[ks].y = p[1];
    }

    v8f c = {};
#pragma unroll
    for (int ks = 0; ks < 8; ++ks) {
      // D = A*B + C, fp32, 16x16x4
      c = __builtin_amdgcn_wmma_f32_16x16x4_f32(
          /*neg_a=*/false, a[ks], /*neg_b=*/false, b[ks],
          /*c_mod=*/(short)0, c, /*reuse_a=*/false, /*reuse_b=*/false);
    }

    const float ysc = ysq[m0 + lq];   // this lane's column = lq

    // D layout: lane (half,lq), vgpr v -> element (row = v + 8*half, col = lq)
#pragma unroll
    for (int v = 0; v < 8; ++v) {
      const int row = n0 + v + 8 * half;
      const float g   = c[v];
      const float c2  = xsv[v] + ysc - 2.0f * g;
      const float Cnm = __builtin_amdgcn_sqrtf(fmaxf(c2, 0.0f));
      const float Knm = __expf(-Cnm * (1.0f / EPS_));
      // r read exactly once -> non-temporal so it doesn't evict x/y tiles
      const float rnm =
          __builtin_nontemporal_load(r + row * M_ + (m0 + lq));
      const float kr = Knm * rnm;
      accK[v]   += Knm;
      accKr[v]  += kr;
      accKrC[v] += kr * Cnm;
    }
  }

  // single cross-lane reduction over the 16 lanes of each half
  // (xor masks < 16 stay within the half)
#pragma unroll
  for (int v = 0; v < 8; ++v) {
    float sk = accK[v], skr = accKr[v], skc = accKrC[v];
#pragma unroll
    for (int m = 1; m < 16; m <<= 1) {
      sk  += __shfl_xor(sk,  m, 32);
      skr += __shfl_xor(skr, m, 32);
      skc += __shfl_xor(skc, m, 32);
    }
    accK[v] = sk; accKr[v] = skr; accKrC[v] = skc;
  }

  if (lq == 0) {  // lanes 0 and 16 each own 8 rows
#pragma unroll
    for (int v = 0; v < 8; ++v) {
      const int row = n0 + v + 8 * half;
      partK[chunk * N_ + row]   = accK[v];
      partKr[chunk * N_ + row]  = accKr[v];
      partKrC[chunk * N_ + row] = accKrC[v];
    }
  }
}

// ---------------------------------------------------------------- kernel 2
// Scalar Sinkhorn recurrence per row + global early-stop, then output.
// Single block, 256 threads, 16 rows/thread (strided for coalescing).
__global__ void sinkhorn_iter_kernel(const float* __restrict__ partK,
                                     const float* __restrict__ partKr,
                                     const float* __restrict__ partKrC,
                                     float* __restrict__ out) {
  const int tid = threadIdx.x;
  const int ROWS = N_ / 256;  // 16

  __shared__ float red[256];
  __shared__ int doneflag;
  if (tid == 0) doneflag = 0;

  float A[ROWS], S[ROWS], T[ROWS], u[ROWS];
#pragma unroll
  for (int j = 0; j < ROWS; ++j) {
    const int row = tid + j * 256;
    float a = 0.f, s = 0.f, t = 0.f;
#pragma unroll
    for (int ch = 0; ch < MCHUNKS_; ++ch) {
      a += partK[ch * N_ + row];
      s += partKr[ch * N_ + row];
      t += partKrC[ch * N_ + row];
    }
    A[j] = a; S[j] = s; T[j] = t;
    u[j] = 1.0f;
  }
  __syncthreads();

  for (int it = 0; it < MAX_ITER_; ++it) {
    float local = 0.f;
#pragma unroll
    for (int j = 0; j < ROWS; ++j) {
      // iter 0: v == 1 so sum(K*v) == A; afterwards v = r/(u+1e-8) -> S/(u+1e-8)
      float un = (it == 0) ? (1.0f / (A[j] + 1e-8f))
                           : (1.0f / (S[j] / (u[j] + 1e-8f) + 1e-8f));
      local += fabsf(un - u[j]);
      u[j] = un;
    }
    red[tid] = local;
    __syncthreads();
    for (int s = 128; s > 0; s >>= 1) {
      if (tid < s) red[tid] += red[tid + s];
      __syncthreads();
    }
    if (tid == 0) {
      const float err = red[0] * (1.0f / (float)N_);
      if (err < THRESH_) doneflag = 1;
    }
    __syncthreads();
    if (doneflag) break;  // uniform: matches reference freeze-after-done
  }

  // out[n] = sum_m u*K*v*C = u/(u+1e-8) * T   (v = r/(u+1e-8), same final u)
#pragma unroll
  for (int j = 0; j < ROWS; ++j) {
    const int row = tid + j * 256;
    out[row] = (u[j] / (u[j] + 1e-8f)) * T[j];
  }
}

// ---------------------------------------------------------------- launcher
extern "C" void kernel_launch(void* const* d_in, const int* in_sizes, int n_in,
                              void* d_out, int out_size, void* d_ws, size_t ws_size,
                              hipStream_t stream) {
  const float* x = (const float*)d_in[0];  // [4096,32]
  const float* y = (const float*)d_in[1];  // [512,32]
  const float* r = (const float*)d_in[2];  // [4096,512]
  float* out = (float*)d_out;              // [4096]

  float* ws      = (float*)d_ws;
  float* xsq     = ws;                       // 4096
  float* ysq     = xsq + N_;                 // 512
  float* partK   = ysq + M_;                 // 4*4096
  float* partKr  = partK + MCHUNKS_ * N_;    // 4*4096
  float* partKrC = partKr + MCHUNKS_ * N_;   // 4*4096
  (void)in_sizes; (void)n_in; (void)out_size; (void)ws_size;

  sq_norms_kernel<<<(N_ + 255) / 256, 256, 0, stream>>>(x, y, xsq, ysq);

  dim3 grid(32, MCHUNKS_);   // 32 blocks * 8 waves = 256 n-tiles ; 4 m-chunks
  gram_rowsum_kernel<<<grid, 256, 0, stream>>>(x, y, r, xsq, ysq,
                                               partK, partKr, partKrC);

  sinkhorn_iter_kernel<<<1, 256, 0, stream>>>(partK, partKr, partKrC, out);
}